// RouteGNN_67585605370083
// MI455X (gfx1250) — compile-verified
//
#include <hip/hip_runtime.h>
#include <math.h>

#define N_NODES 50000
#define N_EDGES 800000
#define ETOT    (N_EDGES + N_NODES)   // 850000, self-loops appended
#define F_IN    128
#define C_HID   64
#define G_GRAPHS 256
#define NEG_SLOPE 0.2f

typedef float v2f __attribute__((ext_vector_type(2)));
typedef float v8f __attribute__((ext_vector_type(8)));

// ---------- small helpers ----------
__device__ __forceinline__ float lrelu(float x) { return x > 0.f ? x : NEG_SLOPE * x; }

// monotonic unsigned encoding of float for atomicMax-based segment max
__device__ __forceinline__ unsigned ordEnc(float f) {
    unsigned u = __float_as_uint(f);
    return (u & 0x80000000u) ? ~u : (u | 0x80000000u);
}
__device__ __forceinline__ float ordDec(unsigned u) {
    unsigned b = (u & 0x80000000u) ? (u & 0x7fffffffu) : ~u;
    return __uint_as_float(b);
}

__global__ void fill_f32(float* p, float v, int n) {
    int i = blockIdx.x * blockDim.x + threadIdx.x;
    if (i < n) p[i] = v;
}
__global__ void fill_u32(unsigned* p, unsigned v, int n) {
    int i = blockIdx.x * blockDim.x + threadIdx.x;
    if (i < n) p[i] = v;
}

// ---------- mean(edge_attr) reduction ----------
__global__ __launch_bounds__(256) void reduce_sum(const float* __restrict__ a, int n, float* out) {
    __shared__ float s[256];
    int tid = threadIdx.x;
    float acc = 0.f;
    for (int i = blockIdx.x * blockDim.x + tid; i < n; i += gridDim.x * blockDim.x)
        acc += a[i];
    s[tid] = acc;
    __syncthreads();
    for (int t = 128; t > 0; t >>= 1) {
        if (tid < t) s[tid] += s[tid + t];
        __syncthreads();
    }
    if (tid == 0) atomicAdd(out, s[0]);
}

// sc[0]=sum(edge_attr) (already accumulated); compute sc[1]=mean, sc[2..3]=k_h = dot(We[h], att_e[h])
__global__ void prep_scalars(const float* __restrict__ We, const float* __restrict__ att_e, float* sc) {
    if (threadIdx.x == 0 && blockIdx.x == 0) {
        sc[1] = sc[0] / (float)N_EDGES;
        float k0 = 0.f, k1 = 0.f;
        for (int c = 0; c < C_HID; ++c) {
            k0 += We[c] * att_e[c];
            k1 += We[C_HID + c] * att_e[C_HID + c];
        }
        sc[2] = k0;
        sc[3] = k1;
    }
}

// ---------- GEMM via fp32 WMMA: D[M,128] = A[M,K] @ W[K,128], row-major ----------
// grid.x = M/16 row tiles; 8 waves per block each own one 16-col tile.
__global__ __launch_bounds__(256) void gemm_wmma_f32(const float* __restrict__ A,
                                                     const float* __restrict__ W,
                                                     float* __restrict__ D, int K) {
    const int wave = threadIdx.x >> 5;
    const int lane = threadIdx.x & 31;
    const int hi   = lane >> 4;       // 0: lanes 0-15, 1: lanes 16-31
    const int lo   = lane & 15;
    const int row0 = blockIdx.x * 16;
    const int col0 = wave * 16;
    const float* Arow = A + (size_t)(row0 + lo) * K;   // A lanes hold M = lane&15
    v8f acc = {0.f, 0.f, 0.f, 0.f, 0.f, 0.f, 0.f, 0.f};
    for (int k0 = 0; k0 < K; k0 += 4) {
        v2f a, b;
        // A 16x4 fp32 layout: VGPR v, lane L -> A[L&15][k0 + v + 2*(L>>4)]
        a.x = Arow[k0 + 0 + 2 * hi];
        a.y = Arow[k0 + 1 + 2 * hi];
        // B 4x16 fp32 layout: VGPR v, lane L -> B[k0 + v + 2*(L>>4)][col0 + (L&15)]
        b.x = W[(size_t)(k0 + 0 + 2 * hi) * 128 + col0 + lo];
        b.y = W[(size_t)(k0 + 1 + 2 * hi) * 128 + col0 + lo];
        acc = __builtin_amdgcn_wmma_f32_16x16x4_f32(false, a, false, b, (short)0, acc, false, false);
    }
    // C/D 16x16 fp32: VGPR r, lane L -> D[r + 8*(L>>4)][L&15]
#pragma unroll
    for (int r = 0; r < 8; ++r)
        D[(size_t)(row0 + r + 8 * hi) * 128 + col0 + lo] = acc[r];
}

// ---------- per-node attention logits: a_src[n,h], a_dst[n,h] ----------
__global__ __launch_bounds__(64) void node_att(const float* __restrict__ xp,
                                               const float* __restrict__ att_s,
                                               const float* __restrict__ att_d,
                                               float* __restrict__ a_src,
                                               float* __restrict__ a_dst) {
    __shared__ float s0[64], s1[64], d0[64], d1[64];
    const int n = blockIdx.x, c = threadIdx.x;
    const float xa = xp[(size_t)n * 128 + c];
    const float xb = xp[(size_t)n * 128 + 64 + c];
    s0[c] = xa * att_s[c];
    s1[c] = xb * att_s[64 + c];
    d0[c] = xa * att_d[c];
    d1[c] = xb * att_d[64 + c];
    __syncthreads();
    for (int t = 32; t > 0; t >>= 1) {
        if (c < t) { s0[c] += s0[c + t]; s1[c] += s1[c + t]; d0[c] += d0[c + t]; d1[c] += d1[c + t]; }
        __syncthreads();
    }
    if (c == 0) {
        a_src[n * 2 + 0] = s0[0];
        a_src[n * 2 + 1] = s1[0];
        a_dst[n * 2 + 0] = d0[0];
        a_dst[n * 2 + 1] = d1[0];
    }
}

__device__ __forceinline__ void edge_decode(const int* __restrict__ ei, const float* __restrict__ eattr,
                                            const float* __restrict__ sc, int e,
                                            int& s, int& d, float& ea) {
    if (e < N_EDGES) {
        s = ei[e];
        d = ei[N_EDGES + e];
        ea = eattr[e];
    } else {
        s = d = e - N_EDGES;   // self-loop
        ea = sc[1];            // mean(edge_attr) fill
    }
}

// ---------- edge pass 1: segment max of leaky_relu(alpha) over dst ----------
__global__ __launch_bounds__(256) void edge_amax(const int* __restrict__ ei, const float* __restrict__ eattr,
                                                 const float* __restrict__ a_src, const float* __restrict__ a_dst,
                                                 const float* __restrict__ sc, unsigned* __restrict__ amax) {
    int e = blockIdx.x * blockDim.x + threadIdx.x;
    if (e >= ETOT) return;
    int s, d; float ea;
    edge_decode(ei, eattr, sc, e, s, d, ea);
    float al0 = lrelu(a_src[s * 2 + 0] + a_dst[d * 2 + 0] + ea * sc[2]);
    float al1 = lrelu(a_src[s * 2 + 1] + a_dst[d * 2 + 1] + ea * sc[3]);
    atomicMax(&amax[d * 2 + 0], ordEnc(al0));
    atomicMax(&amax[d * 2 + 1], ordEnc(al1));
}

// ---------- edge pass 2: ex = exp(alpha - amax[dst]), denom = segsum(ex) ----------
__global__ __launch_bounds__(256) void edge_exp(const int* __restrict__ ei, const float* __restrict__ eattr,
                                                const float* __restrict__ a_src, const float* __restrict__ a_dst,
                                                const float* __restrict__ sc, const unsigned* __restrict__ amax,
                                                float* __restrict__ ex, float* __restrict__ den) {
    int e = blockIdx.x * blockDim.x + threadIdx.x;
    if (e >= ETOT) return;
    int s, d; float ea;
    edge_decode(ei, eattr, sc, e, s, d, ea);
    float al0 = lrelu(a_src[s * 2 + 0] + a_dst[d * 2 + 0] + ea * sc[2]);
    float al1 = lrelu(a_src[s * 2 + 1] + a_dst[d * 2 + 1] + ea * sc[3]);
    float e0 = __expf(al0 - ordDec(amax[d * 2 + 0]));
    float e1 = __expf(al1 - ordDec(amax[d * 2 + 1]));
    ex[e * 2 + 0] = e0;
    ex[e * 2 + 1] = e1;
    atomicAdd(&den[d * 2 + 0], e0);
    atomicAdd(&den[d * 2 + 1], e1);
}

// ---------- edge pass 3: agg[dst, :] += (ex/den) * xp[src, :]  (128 chans/edge) ----------
__global__ __launch_bounds__(256) void edge_scatter(const int* __restrict__ ei,
                                                    const float* __restrict__ ex,
                                                    const float* __restrict__ den,
                                                    const float* __restrict__ xp,
                                                    float* __restrict__ agg) {
    long long id = (long long)blockIdx.x * blockDim.x + threadIdx.x;
    if (id >= (long long)ETOT * 128) return;
    int e = (int)(id >> 7);
    int c = (int)(id & 127);
    int s, d;
    if (e < N_EDGES) { s = ei[e]; d = ei[N_EDGES + e]; }
    else             { s = d = e - N_EDGES; }
    int h = c >> 6;
    float coef = ex[e * 2 + h] / den[d * 2 + h];
    atomicAdd(&agg[(size_t)d * 128 + c], coef * xp[(size_t)s * 128 + c]);
}

// ---------- head mean + bias + relu: h[n,c] = relu(0.5*(agg[n,c]+agg[n,64+c]) + b[c]) ----------
__global__ __launch_bounds__(256) void head_mean_relu(const float* __restrict__ agg,
                                                      const float* __restrict__ bias,
                                                      float* __restrict__ h) {
    int i = blockIdx.x * blockDim.x + threadIdx.x;
    if (i >= N_NODES * C_HID) return;
    int n = i >> 6, c = i & 63;
    float v = 0.5f * (agg[(size_t)n * 128 + c] + agg[(size_t)n * 128 + 64 + c]) + bias[c];
    h[i] = fmaxf(v, 0.f);
}

// ---------- global mean pool (accumulate) ----------
__global__ __launch_bounds__(64) void pool_kernel(const float* __restrict__ h, const int* __restrict__ batch,
                                                  float* __restrict__ pool, float* __restrict__ cnt) {
    int n = blockIdx.x, c = threadIdx.x;
    int g = batch[n];
    atomicAdd(&pool[(size_t)g * C_HID + c], h[(size_t)n * C_HID + c]);
    if (c == 0) atomicAdd(&cnt[g], 1.0f);
}

// ---------- final: out[g] = dot(pool[g]/max(cnt,1), lin_w) + lin_b ----------
__global__ __launch_bounds__(64) void final_kernel(const float* __restrict__ pool, const float* __restrict__ cnt,
                                                   const float* __restrict__ lin_w, const float* __restrict__ lin_b,
                                                   float* __restrict__ out) {
    __shared__ float s[64];
    int g = blockIdx.x, c = threadIdx.x;
    float cc = fmaxf(cnt[g], 1.0f);
    s[c] = (pool[(size_t)g * C_HID + c] / cc) * lin_w[c];
    __syncthreads();
    for (int t = 32; t > 0; t >>= 1) {
        if (c < t) s[c] += s[c + t];
        __syncthreads();
    }
    if (c == 0) out[g] = s[0] + lin_b[0];
}

// =================== launcher ===================
extern "C" void kernel_launch(void* const* d_in, const int* in_sizes, int n_in,
                              void* d_out, int out_size, void* d_ws, size_t ws_size,
                              hipStream_t stream) {
    const float* x      = (const float*)d_in[0];
    const int*   ei     = (const int*)d_in[1];     // [2,E] row-major: src then dst
    const float* eattr  = (const float*)d_in[2];   // [E,1]
    const int*   batch  = (const int*)d_in[3];
    const float* W1     = (const float*)d_in[4];
    const float* asrc1  = (const float*)d_in[5];
    const float* adst1  = (const float*)d_in[6];
    const float* We1    = (const float*)d_in[7];
    const float* ae1    = (const float*)d_in[8];
    const float* b1     = (const float*)d_in[9];
    const float* W2     = (const float*)d_in[10];
    const float* asrc2  = (const float*)d_in[11];
    const float* adst2  = (const float*)d_in[12];
    const float* We2    = (const float*)d_in[13];
    const float* ae2    = (const float*)d_in[14];
    const float* b2     = (const float*)d_in[15];
    const float* lin_w  = (const float*)d_in[16];
    const float* lin_b  = (const float*)d_in[17];
    float* out = (float*)d_out;

    // workspace layout (floats)
    float* ws   = (float*)d_ws;
    float* XP   = ws;                        // N*128
    float* AGG  = XP  + (size_t)N_NODES * 128;
    float* HB   = AGG + (size_t)N_NODES * 128;  // N*64
    float* EX   = HB  + (size_t)N_NODES * 64;   // ETOT*2
    float* ASRC = EX  + (size_t)ETOT * 2;       // N*2
    float* ADST = ASRC + (size_t)N_NODES * 2;
    unsigned* AMAX = (unsigned*)(ADST + (size_t)N_NODES * 2); // N*2
    float* DEN  = (float*)AMAX + (size_t)N_NODES * 2;
    float* POOL = DEN + (size_t)N_NODES * 2;    // G*64
    float* CNT  = POOL + (size_t)G_GRAPHS * 64; // G
    float* SC   = CNT + G_GRAPHS;               // 8 scalars

    const unsigned NEG_INF_ORD = 0x007FFFFFu;   // ordEnc(-inf)
    const int B = 256;
    const int gEdges = (ETOT + B - 1) / B;
    const long long scatN = (long long)ETOT * 128;
    const int gScat = (int)((scatN + B - 1) / B);
    const int gNC = (N_NODES * C_HID + B - 1) / B;

    // scalars: sum/mean of edge_attr
    fill_f32<<<1, 8, 0, stream>>>(SC, 0.f, 8);
    reduce_sum<<<256, 256, 0, stream>>>(eattr, N_EDGES, &SC[0]);

    for (int layer = 0; layer < 2; ++layer) {
        const float* A    = (layer == 0) ? x     : HB;
        const int    K    = (layer == 0) ? F_IN  : C_HID;
        const float* W    = (layer == 0) ? W1    : W2;
        const float* asrc = (layer == 0) ? asrc1 : asrc2;
        const float* adst = (layer == 0) ? adst1 : adst2;
        const float* We   = (layer == 0) ? We1   : We2;
        const float* ae   = (layer == 0) ? ae1   : ae2;
        const float* bias = (layer == 0) ? b1    : b2;

        prep_scalars<<<1, 1, 0, stream>>>(We, ae, SC);
        gemm_wmma_f32<<<N_NODES / 16, 256, 0, stream>>>(A, W, XP, K);
        node_att<<<N_NODES, 64, 0, stream>>>(XP, asrc, adst, ASRC, ADST);

        fill_u32<<<(N_NODES * 2 + B - 1) / B, B, 0, stream>>>(AMAX, NEG_INF_ORD, N_NODES * 2);
        fill_f32<<<(N_NODES * 2 + B - 1) / B, B, 0, stream>>>(DEN, 0.f, N_NODES * 2);
        fill_f32<<<(N_NODES * 128 + B - 1) / B, B, 0, stream>>>(AGG, 0.f, N_NODES * 128);

        edge_amax<<<gEdges, B, 0, stream>>>(ei, eattr, ASRC, ADST, SC, AMAX);
        edge_exp<<<gEdges, B, 0, stream>>>(ei, eattr, ASRC, ADST, SC, AMAX, EX, DEN);
        edge_scatter<<<gScat, B, 0, stream>>>(ei, EX, DEN, XP, AGG);
        head_mean_relu<<<gNC, B, 0, stream>>>(AGG, bias, HB);
    }

    // global mean pool + final linear
    fill_f32<<<(G_GRAPHS * C_HID + B - 1) / B, B, 0, stream>>>(POOL, 0.f, G_GRAPHS * C_HID);
    fill_f32<<<1, G_GRAPHS, 0, stream>>>(CNT, 0.f, G_GRAPHS);
    pool_kernel<<<N_NODES, 64, 0, stream>>>(HB, batch, POOL, CNT);
    final_kernel<<<G_GRAPHS, 64, 0, stream>>>(POOL, CNT, lin_w, lin_b, out);
}